// HeteroGAT_67095979098387
// MI455X (gfx1250) — compile-verified
//
#include <hip/hip_runtime.h>
#include <hip/hip_bf16.h>
#include <math.h>

#define DIN 128
#define DOUT 128
#define SLOPE 0.2f

typedef float v2f __attribute__((ext_vector_type(2)));
typedef float v8f __attribute__((ext_vector_type(8)));

// ---------------------------------------------------------------------------
// init: zero output accumulator, m = -inf, denom = 0
// ---------------------------------------------------------------------------
__global__ void init_k(float* __restrict__ out, float* __restrict__ m,
                       float* __restrict__ denom, int n) {
    int i = blockIdx.x * blockDim.x + threadIdx.x;
    int total = n * DOUT;
    if (i < total) out[i] = 0.0f;
    if (i < n) {
        m[i] = -INFINITY;
        denom[i] = 0.0f;
    }
}

// ---------------------------------------------------------------------------
// h = x @ W via V_WMMA_F32_16X16X4_F32 (f32-exact tensor path).
// Block = 256 threads = 8 wave32s. Block covers 16 rows x 128 cols; each wave
// owns one 16-col tile and accumulates K=128 in 32 steps of K=4.
// A-frag (16x4): lanes 0-15 row M=lane holds K{0,1}; lanes 16-31 hold K{2,3}.
// B-frag (4x16): V0 = rows K0(K2), V1 = rows K1(K3), cols striped on lanes.
// C/D: VGPR r -> row r (lanes 0-15) / row r+8 (lanes 16-31).
// ---------------------------------------------------------------------------
__global__ void __launch_bounds__(256)
gemm_wmma(const float* __restrict__ X, const float* __restrict__ W,
          float* __restrict__ H, int nrows) {
    __shared__ float xs[16][DIN + 4];   // +4 pad -> conflict-free column reads
    const int row0 = blockIdx.x * 16;
    const int tid  = threadIdx.x;

    // cooperative stage of the 16x128 x-tile into LDS
    for (int i = tid; i < 16 * DIN; i += 256) {
        int r = i >> 7, c = i & (DIN - 1);
        int gr = row0 + r;
        xs[r][c] = (gr < nrows) ? X[(size_t)gr * DIN + c] : 0.0f;
    }
    __syncthreads();

    const int wave = tid >> 5;          // 0..7 -> column tile
    const int lane = tid & 31;
    const int half = lane >> 4;         // 0 or 1
    const int l16  = lane & 15;
    const int col0 = wave * 16;

    v8f acc = {};
#pragma unroll
    for (int k = 0; k < DIN / 4; ++k) {
        const int kk = k * 4 + half * 2;
        v2f a, b;
        a.x = xs[l16][kk];
        a.y = xs[l16][kk + 1];
        b.x = W[(size_t)kk * DOUT + col0 + l16];
        b.y = W[(size_t)(kk + 1) * DOUT + col0 + l16];
        acc = __builtin_amdgcn_wmma_f32_16x16x4_f32(
            /*neg_a=*/false, a, /*neg_b=*/false, b,
            /*c_mod=*/(short)0, acc, /*reuse_a=*/false, /*reuse_b=*/false);
    }

    const int orow = row0 + half * 8;
#pragma unroll
    for (int r = 0; r < 8; ++r) {
        int gr = orow + r;
        if (gr < nrows) H[(size_t)gr * DOUT + col0 + l16] = acc[r];
    }
}

// ---------------------------------------------------------------------------
// a_src[i] = h[i] . att_src ; a_dst[i] = h[i] . att_dst   (wave per row)
// ---------------------------------------------------------------------------
__global__ void att_dots(const float* __restrict__ H,
                         const float* __restrict__ att_s,
                         const float* __restrict__ att_d,
                         float* __restrict__ aS, float* __restrict__ aD, int n) {
    int w = (int)((blockIdx.x * (size_t)blockDim.x + threadIdx.x) >> 5);
    int lane = threadIdx.x & 31;
    if (w >= n) return;
    float4 h4 = ((const float4*)(H + (size_t)w * DOUT))[lane];
    float4 s4 = ((const float4*)att_s)[lane];
    float4 d4 = ((const float4*)att_d)[lane];
    float ps = h4.x * s4.x + h4.y * s4.y + h4.z * s4.z + h4.w * s4.w;
    float pd = h4.x * d4.x + h4.y * d4.y + h4.z * d4.z + h4.w * d4.w;
#pragma unroll
    for (int off = 16; off > 0; off >>= 1) {
        ps += __shfl_xor(ps, off, 32);
        pd += __shfl_xor(pd, off, 32);
    }
    if (lane == 0) { aS[w] = ps; aD[w] = pd; }
}

// float atomic max that is correct for mixed signs (init must be -inf)
__device__ __forceinline__ void atomicMaxF(float* addr, float v) {
    if (v >= 0.0f) atomicMax((int*)addr, __float_as_int(v));
    else           atomicMin((unsigned int*)addr, __float_as_uint(v));
}

__device__ __forceinline__ void edge_sd(const long long* __restrict__ ei,
                                        int i, int E, int& s, int& d) {
    if (i < E) { s = (int)ei[i]; d = (int)ei[E + i]; }
    else       { s = d = i - E; }            // self-loop
}

__device__ __forceinline__ float leaky(float e) {
    return e > 0.0f ? e : SLOPE * e;
}

// ---------------------------------------------------------------------------
// segment max of leaky_relu(a_src[s] + a_dst[d]) over dst
// ---------------------------------------------------------------------------
__global__ void edge_max(const long long* __restrict__ ei,
                         const float* __restrict__ aS,
                         const float* __restrict__ aD,
                         float* __restrict__ m, int E, int N) {
    int i = blockIdx.x * blockDim.x + threadIdx.x;
    if (i >= E + N) return;
    int s, d; edge_sd(ei, i, E, s, d);
    atomicMaxF(&m[d], leaky(aS[s] + aD[d]));
}

// ---------------------------------------------------------------------------
// denom[d] += exp(e - m[d])
// ---------------------------------------------------------------------------
__global__ void edge_expsum(const long long* __restrict__ ei,
                            const float* __restrict__ aS,
                            const float* __restrict__ aD,
                            const float* __restrict__ m,
                            float* __restrict__ denom, int E, int N) {
    int i = blockIdx.x * blockDim.x + threadIdx.x;
    if (i >= E + N) return;
    int s, d; edge_sd(ei, i, E, s, d);
    float e = leaky(aS[s] + aD[d]);
    atomicAdd(&denom[d], __expf(e - m[d]));
}

// ---------------------------------------------------------------------------
// out[d] += alpha * h[s]   (wave per edge; h gathers hit the 192MB L2)
// ---------------------------------------------------------------------------
__global__ void edge_scatter(const long long* __restrict__ ei,
                             const float* __restrict__ aS,
                             const float* __restrict__ aD,
                             const float* __restrict__ m,
                             const float* __restrict__ denom,
                             const float* __restrict__ H,
                             float* __restrict__ out, int E, int N) {
    int w = (int)((blockIdx.x * (size_t)blockDim.x + threadIdx.x) >> 5);
    int lane = threadIdx.x & 31;
    if (w >= E + N) return;
    int s, d; edge_sd(ei, w, E, s, d);
    float e = leaky(aS[s] + aD[d]);
    float alpha = __expf(e - m[d]) / denom[d];
    float4 h4 = ((const float4*)(H + (size_t)s * DOUT))[lane];
    float* o = out + (size_t)d * DOUT + lane * 4;
    atomicAdd(o + 0, alpha * h4.x);
    atomicAdd(o + 1, alpha * h4.y);
    atomicAdd(o + 2, alpha * h4.z);
    atomicAdd(o + 3, alpha * h4.w);
}

// ---------------------------------------------------------------------------
// out = relu(out + bias)
// ---------------------------------------------------------------------------
__global__ void bias_relu(float* __restrict__ out, const float* __restrict__ b,
                          int n) {
    int i = blockIdx.x * blockDim.x + threadIdx.x;
    int total = n * DOUT;
    if (i >= total) return;
    float v = out[i] + b[i & (DOUT - 1)];
    out[i] = v > 0.0f ? v : 0.0f;
}

// ---------------------------------------------------------------------------
extern "C" void kernel_launch(void* const* d_in, const int* in_sizes, int n_in,
                              void* d_out, int out_size, void* d_ws, size_t ws_size,
                              hipStream_t stream) {
    const int N = in_sizes[0] / DIN;   // 50000
    const int E = in_sizes[1] / 2;     // 800000
    const int T = E + N;

    float* ws = (float*)d_ws;
    const size_t perType = (size_t)N * DOUT + 4 * (size_t)N;

    for (int t = 0; t < 2; ++t) {
        const float*     x  = (const float*)d_in[t * 6 + 0];
        const long long* ei = (const long long*)d_in[t * 6 + 1];
        const float*     W  = (const float*)d_in[t * 6 + 2];
        const float*     as = (const float*)d_in[t * 6 + 3];
        const float*     ad = (const float*)d_in[t * 6 + 4];
        const float*     b  = (const float*)d_in[t * 6 + 5];

        float* out = (float*)d_out + (size_t)t * N * DOUT;
        float* H   = ws + (size_t)t * perType;
        float* aS  = H + (size_t)N * DOUT;
        float* aD  = aS + N;
        float* m   = aD + N;
        float* dn  = m + N;

        init_k<<<(N * DOUT + 255) / 256, 256, 0, stream>>>(out, m, dn, N);
        gemm_wmma<<<(N + 15) / 16, 256, 0, stream>>>(x, W, H, N);
        att_dots<<<((size_t)N * 32 + 255) / 256, 256, 0, stream>>>(H, as, ad, aS, aD, N);
        edge_max<<<(T + 255) / 256, 256, 0, stream>>>(ei, aS, aD, m, E, N);
        edge_expsum<<<(T + 255) / 256, 256, 0, stream>>>(ei, aS, aD, m, dn, E, N);
        edge_scatter<<<((size_t)T * 32 + 255) / 256, 256, 0, stream>>>(ei, aS, aD, m, dn, H, out, E, N);
        bias_relu<<<(N * DOUT + 255) / 256, 256, 0, stream>>>(out, b, N);
    }
}